// RankedListLoss_59614146069099
// MI455X (gfx1250) — compile-verified
//
#include <hip/hip_runtime.h>

typedef __attribute__((ext_vector_type(2))) float v2f;
typedef __attribute__((ext_vector_type(8))) float v8f;

#define N_EMB   8192
#define DIM     128
#define NCLS    64
#define ALPHA   1.2f
#define POS_TH  0.8f      /* ALPHA - MARGIN */
#define TPARAM  10.0f

#define TILE_M  16
#define NBLK    (N_EMB / TILE_M)   /* 512 */
#define WAVES   8
#define BLOCK   (WAVES * 32)

// ---------------- Kernel 1: row squared norms -----------------------------
__global__ void rll_sq(const float* __restrict__ emb, float* __restrict__ sq) {
    int wave = (blockIdx.x * blockDim.x + threadIdx.x) >> 5;
    int lane = threadIdx.x & 31;
    if (wave >= N_EMB) return;
    const float* row = emb + (size_t)wave * DIM;
    float s = 0.f;
    #pragma unroll
    for (int k = 0; k < DIM / 32; ++k) {
        float v = row[lane + 32 * k];
        s += v * v;
    }
    #pragma unroll
    for (int off = 16; off > 0; off >>= 1) s += __shfl_xor(s, off, 32);
    if (lane == 0) sq[wave] = s;
}

// ---------------- Kernel 2: fused GEMM + masks + reductions ---------------
__global__ __launch_bounds__(BLOCK) void rll_main(
        const float* __restrict__ emb, const int* __restrict__ labels,
        const float* __restrict__ sq,
        float* __restrict__ row_wsum, float* __restrict__ row_num,
        float* __restrict__ pos_part) {
    __shared__ float lds_wsum[WAVES][TILE_M];
    __shared__ float lds_num[WAVES][TILE_M];
    __shared__ float lds_pos[WAVES];

    const int lane = threadIdx.x & 31;
    const int wave = threadIdx.x >> 5;
    const int half = lane >> 4;       // 0: K pair {0,1}, 1: K pair {2,3}
    const int l16  = lane & 15;
    const int i_base = blockIdx.x * TILE_M;

    // Preload A fragments for all 32 K-steps (16 rows x 128 cols strip).
    // f32 16x16x4 A layout: lane m<16 holds (M=m, K=0..1), lane m+16 holds (M=m, K=2..3).
    const float* arow = emb + (size_t)(i_base + l16) * DIM + 2 * half;
    v2f a[32];
    #pragma unroll
    for (int t = 0; t < 32; ++t)
        a[t] = *(const v2f*)(arow + 4 * t);

    // Per-row (M) constants; C/D layout: c[v] is (M = v + 8*half, N = l16).
    float sqI[8]; int labI[8];
    #pragma unroll
    for (int v = 0; v < 8; ++v) {
        int m = v + 8 * half;
        sqI[v]  = sq[i_base + m];
        labI[v] = labels[i_base + m];
    }

    float pos_acc = 0.f;
    float wsum_acc[8] = {0.f,0.f,0.f,0.f,0.f,0.f,0.f,0.f};
    float num_acc [8] = {0.f,0.f,0.f,0.f,0.f,0.f,0.f,0.f};

    for (int jt = wave; jt < NBLK; jt += WAVES) {
        const int j_base = jt * TILE_M;
        // B fragment load pattern is identical to A (E * E^T symmetry).
        const float* brow = emb + (size_t)(j_base + l16) * DIM + 2 * half;
        // Two independent accumulation chains (K halves) to double WMMA ILP.
        v8f c0 = {0.f,0.f,0.f,0.f,0.f,0.f,0.f,0.f};
        v8f c1 = {0.f,0.f,0.f,0.f,0.f,0.f,0.f,0.f};
        #pragma unroll
        for (int t = 0; t < 16; ++t) {
            v2f b0 = *(const v2f*)(brow + 4 * t);
            v2f b1 = *(const v2f*)(brow + 4 * (t + 16));
            c0 = __builtin_amdgcn_wmma_f32_16x16x4_f32(
                    false, a[t],      false, b0, (short)0, c0, false, false);
            c1 = __builtin_amdgcn_wmma_f32_16x16x4_f32(
                    false, a[t + 16], false, b1, (short)0, c1, false, false);
        }
        const float sqJ  = sq[j_base + l16];
        const int   labJ = labels[j_base + l16];
        const int   jglb = j_base + l16;
        #pragma unroll
        for (int v = 0; v < 8; ++v) {
            const int iglb = i_base + v + 8 * half;
            const float dot  = c0[v] + c1[v];
            const float dist = sqI[v] + sqJ - 2.0f * dot;
            if (labI[v] == labJ) {
                if (iglb != jglb && dist > POS_TH) pos_acc += dist - POS_TH;
            } else if (dist < ALPHA) {
                const float gap = ALPHA - dist;
                const float w = __expf(TPARAM * gap);
                wsum_acc[v] += w;
                num_acc[v]  += gap * w;
            }
        }
    }

    // Column reduction within each 16-lane half (N dimension).
    #pragma unroll
    for (int v = 0; v < 8; ++v) {
        float w = wsum_acc[v], n = num_acc[v];
        #pragma unroll
        for (int off = 1; off < 16; off <<= 1) {
            w += __shfl_xor(w, off, 32);
            n += __shfl_xor(n, off, 32);
        }
        wsum_acc[v] = w; num_acc[v] = n;
    }
    if (l16 == 0) {
        #pragma unroll
        for (int v = 0; v < 8; ++v) {
            lds_wsum[wave][v + 8 * half] = wsum_acc[v];
            lds_num [wave][v + 8 * half] = num_acc[v];
        }
    }
    #pragma unroll
    for (int off = 1; off < 32; off <<= 1) pos_acc += __shfl_xor(pos_acc, off, 32);
    if (lane == 0) lds_pos[wave] = pos_acc;
    __syncthreads();

    // Owner-writes (one WG per row-strip) => deterministic, no atomics.
    if (threadIdx.x < TILE_M) {
        float w = 0.f, n = 0.f;
        #pragma unroll
        for (int q = 0; q < WAVES; ++q) { w += lds_wsum[q][threadIdx.x]; n += lds_num[q][threadIdx.x]; }
        row_wsum[i_base + threadIdx.x] = w;
        row_num [i_base + threadIdx.x] = n;
    }
    if (threadIdx.x == 0) {
        float p = 0.f;
        #pragma unroll
        for (int q = 0; q < WAVES; ++q) p += lds_pos[q];
        pos_part[blockIdx.x] = p;
    }
}

// ---------------- Kernel 3: class segment-sum + final loss ----------------
__global__ void rll_final(const int* __restrict__ labels,
                          const float* __restrict__ row_wsum,
                          const float* __restrict__ row_num,
                          const float* __restrict__ pos_part,
                          float* __restrict__ out) {
    __shared__ float cls_w[NCLS], cls_n[NCLS], red[256];
    const int t = threadIdx.x;
    if (t < NCLS) {
        float w = 0.f, n = 0.f;
        for (int i = 0; i < N_EMB; ++i) {
            if (labels[i] == t) { w += row_wsum[i]; n += row_num[i]; }
        }
        cls_w[t] = w; cls_n[t] = n;
    }
    red[t] = pos_part[t] + pos_part[t + 256];   // NBLK == 512
    __syncthreads();
    if (t == 0) {
        float pos = 0.f;
        for (int q = 0; q < 256; ++q) pos += red[q];
        float neg = 0.f;
        for (int c = 0; c < NCLS; ++c)
            if (cls_w[c] > 0.f) neg += cls_n[c] / cls_w[c];
        out[0] = pos + neg;
        out[1] = (float)N_EMB;
    }
}

extern "C" void kernel_launch(void* const* d_in, const int* in_sizes, int n_in,
                              void* d_out, int out_size, void* d_ws, size_t ws_size,
                              hipStream_t stream) {
    const float* emb    = (const float*)d_in[0];
    const int*   labels = (const int*)d_in[1];
    float* out = (float*)d_out;
    float* ws  = (float*)d_ws;

    float* sq       = ws;                // N_EMB
    float* row_wsum = ws + N_EMB;        // N_EMB
    float* row_num  = ws + 2 * N_EMB;    // N_EMB
    float* pos_part = ws + 3 * N_EMB;    // NBLK (512)

    rll_sq  <<<N_EMB / 8, 256, 0, stream>>>(emb, sq);
    rll_main<<<NBLK,      BLOCK, 0, stream>>>(emb, labels, sq, row_wsum, row_num, pos_part);
    rll_final<<<1,        256,  0, stream>>>(labels, row_wsum, row_num, pos_part, out);
}